// GodunovSolver_83365315215363
// MI455X (gfx1250) — compile-verified
//
#include <hip/hip_runtime.h>
#include <cstdint>

#if __has_builtin(__builtin_amdgcn_tensor_store_from_lds) && __has_builtin(__builtin_amdgcn_s_wait_tensorcnt)
#define USE_TDM 1
#else
#define USE_TDM 0
#endif

namespace {
constexpr int kB        = 16;
constexpr int kP        = 8;
constexpr int kNT       = 512;
constexpr int kNX       = 2048;
constexpr int kCells    = 8;                 // cells per thread
constexpr int kThreads  = kNX / kCells;      // 256 threads = 8 waves (wave32)
constexpr int kWaveCells = 32 * kCells;      // 256 cells = 1KB per wave slice
}

typedef __attribute__((ext_vector_type(4))) unsigned int v4u;
typedef __attribute__((ext_vector_type(4))) int          v4i;
typedef __attribute__((ext_vector_type(8))) int          v8i;

__device__ __forceinline__ float godunov_flux(float uL, float uR) {
  const float fL = uL * (1.0f - uL);
  const float fR = uR * (1.0f - uR);
  const float fmin = fminf(fL, fR);
  const float fmax = ((uL >= 0.5f) && (uR <= 0.5f)) ? 0.25f : fmaxf(fL, fR);
  return (uL <= uR) ? fmin : fmax;
}

#if USE_TDM
// One TDM DMA: nelem f32 from LDS byte offset -> global address (1 x nelem tile).
// D# layout per CDNA5 ISA.
__device__ __forceinline__ void tdm_store_slice(uint32_t lds_byte, uint64_t gaddr,
                                                uint32_t nelem) {
  v4u g0;
  g0[0] = 1u;                                              // count=1, user mode
  g0[1] = lds_byte;                                        // lds_addr [63:32]
  g0[2] = (uint32_t)(gaddr & 0xFFFFFFFFull);               // global_addr lo
  g0[3] = (uint32_t)((gaddr >> 32) & 0x01FFFFFFull)        // global_addr[56:32]
        | (2u << 30);                                      // type=2 ("image")
  v8i g1;
  g1[0] = (int)(2u << 16);                                 // data_size=2 -> 4B; no mcast
  g1[1] = (int)((nelem & 0xFFFFu) << 16);                  // tensor_dim0 lo16
  g1[2] = (int)(((nelem >> 16) & 0xFFFFu) | (1u << 16));   // tensor_dim0 hi16 | tensor_dim1=1
  g1[3] = (int)((nelem & 0xFFFFu) << 16);                  // tensor_dim1 hi=0 | tile_dim0
  g1[4] = 1;                                               // tile_dim1=1, tile_dim2=0
  g1[5] = (int)nelem;                                      // tensor_dim0_stride lo32
  g1[6] = 0;                                               // stride0 hi | stride1 lo
  g1[7] = 0;                                               // stride1 hi
  v4i g2 = {0, 0, 0, 0};                                   // <=2D: groups 2/3 unused
  v4i g3 = {0, 0, 0, 0};
#if defined(__clang_major__) && (__clang_major__ >= 23)
  v8i gx = {0, 0, 0, 0, 0, 0, 0, 0};
  __builtin_amdgcn_tensor_store_from_lds(g0, g1, g2, g3, gx, 0);
#else
  __builtin_amdgcn_tensor_store_from_lds(g0, g1, g2, g3, 0);
#endif
}
#endif

__global__ __launch_bounds__(kThreads)
void godunov_tdm_kernel(const float* __restrict__ xs,    // B x (P+1)
                        const float* __restrict__ ks,    // B x P
                        const int*   __restrict__ mask,  // B x P
                        const float* __restrict__ dx,    // B
                        const float* __restrict__ dt,    // B
                        float* __restrict__ out)         // B x NT x NX
{
  __shared__ float buf[2][kNX];    // double-buffered state row (16 KB of 320 KB)
  __shared__ float ghosts[2];

  const int b   = blockIdx.x;
  const int tid = threadIdx.x;
  const int j   = tid * kCells;

  const float dxv = dx[0];         // reference uses dx[0]/dt[0] globally
  const float lam = dt[0] / dxv;

  // ---- piecewise-constant initial condition ----
  float bnds[kP], kv[kP];
  int np = 0;
  #pragma unroll
  for (int p = 0; p < kP; ++p) {
    const int m = mask[b * kP + p];
    np += (m != 0) ? 1 : 0;
    bnds[p] = (m != 0) ? xs[b * (kP + 1) + p + 1] : __builtin_inff();
    kv[p]   = ks[b * kP + p];
  }

  float u[kCells];
  #pragma unroll
  for (int c = 0; c < kCells; ++c) {
    const float xc = ((float)(j + c) + 0.5f) * dxv;
    int idx = 0;
    #pragma unroll
    for (int p = 0; p < kP; ++p) idx += (xc >= bnds[p]) ? 1 : 0;
    const int mx = np - 1;
    idx = (idx < mx) ? idx : mx;
    float val = kv[0];                       // branchless gather (no v-scratch)
    #pragma unroll
    for (int p = 1; p < kP; ++p) val = (idx >= p) ? kv[p] : val;
    u[c] = val;
  }

  #pragma unroll
  for (int c = 0; c < kCells; ++c) buf[0][j + c] = u[c];
  if (tid == 0)            ghosts[0] = u[0];
  if (tid == kThreads - 1) ghosts[1] = u[kCells - 1];
  __syncthreads();
  const float gL = ghosts[0];
  const float gR = ghosts[1];

  const uint32_t sliceByte = (uint32_t)((tid >> 5) * kWaveCells) * 4u; // wave slice
  const uint64_t outBase   = (uint64_t)(uintptr_t)out
                           + (uint64_t)b * (uint64_t)kNT * (uint64_t)kNX * 4ull;

#if USE_TDM
  const uint32_t lds0 = (uint32_t)(uintptr_t)&buf[0][0];
  const uint32_t lds1 = (uint32_t)(uintptr_t)&buf[1][0];
  // row 0: each wave DMAs its own 1KB slice straight from LDS
  tdm_store_slice(lds0 + sliceByte, outBase + sliceByte, kWaveCells);
#else
  {
    float* o = out + (uint64_t)b * kNT * kNX;
    #pragma unroll
    for (int c = 0; c < kCells; ++c) o[j + c] = u[c];
  }
#endif

  int p = 0;
  for (int t = 1; t < kNT; ++t) {
    // read halo window from current buffer (DMA of this row may still be in
    // flight -- concurrent reads are fine)
    float un[kCells + 2];
    un[0] = (j == 0) ? gL : buf[p][j - 1];
    #pragma unroll
    for (int c = 0; c < kCells; ++c) un[c + 1] = buf[p][j + c];
    un[kCells + 1] = (j + kCells == kNX) ? gR : buf[p][j + kCells];

    float F[kCells + 1];
    #pragma unroll
    for (int c = 0; c <= kCells; ++c) F[c] = godunov_flux(un[c], un[c + 1]);

    float nu[kCells];
    #pragma unroll
    for (int c = 0; c < kCells; ++c) nu[c] = un[c + 1] - lam * (F[c + 1] - F[c]);

#if USE_TDM
    // before overwriting buf[1-p]: our own DMA of that slice (issued at t-2)
    // must be done; allow the t-1 DMA (other buffer) to stay in flight.
    __builtin_amdgcn_s_wait_tensorcnt(1);
#endif
    #pragma unroll
    for (int c = 0; c < kCells; ++c) buf[1 - p][j + c] = nu[c];
    __syncthreads();   // flush DS writes + make row t visible to all waves

#if USE_TDM
    const uint32_t ldsRow = (p == 0) ? lds1 : lds0;
    tdm_store_slice(ldsRow + sliceByte,
                    outBase + (uint64_t)t * (uint64_t)kNX * 4ull + sliceByte,
                    kWaveCells);
#else
    {
      float* o = out + ((uint64_t)b * kNT + (uint64_t)t) * kNX;
      #pragma unroll
      for (int c = 0; c < kCells; ++c) o[j + c] = nu[c];
    }
#endif
    p ^= 1;
  }

#if USE_TDM
  __builtin_amdgcn_s_wait_tensorcnt(0);    // drain DMAs before wave exit
#endif
}

extern "C" void kernel_launch(void* const* d_in, const int* in_sizes, int n_in,
                              void* d_out, int out_size, void* d_ws, size_t ws_size,
                              hipStream_t stream) {
  (void)in_sizes; (void)n_in; (void)d_ws; (void)ws_size; (void)out_size;
  const float* xs   = (const float*)d_in[0];
  const float* ks   = (const float*)d_in[1];
  const int*   mask = (const int*)d_in[2];
  const float* dx   = (const float*)d_in[3];
  const float* dt   = (const float*)d_in[4];
  // d_in[5] (t_coords) only conveys shape; unused.
  float* out = (float*)d_out;
  godunov_tdm_kernel<<<kB, kThreads, 0, stream>>>(xs, ks, mask, dx, dt, out);
}